// ChamferNormalLoss_13091060318819
// MI455X (gfx1250) — compile-verified
//
#include <hip/hip_runtime.h>
#include <hip/hip_bf16.h>
#include <math.h>

typedef __attribute__((ext_vector_type(2))) float v2f;
typedef __attribute__((ext_vector_type(8))) float v8f;

#define CN_EPS 1e-12f

// ---------------------------------------------------------------------------
// Kernel 0: zero the vertex-normal accumulator region + the scalar accumulator
// ---------------------------------------------------------------------------
__global__ void cn_zero_kernel(float* __restrict__ vn, float* __restrict__ acc, int n) {
    int i = blockIdx.x * blockDim.x + threadIdx.x;
    if (i < n) vn[i] = 0.0f;
    if (i == 0) acc[0] = 0.0f;
}

// ---------------------------------------------------------------------------
// Kernel 1: scatter-add face normals into per-vertex normals (gt mesh)
// ---------------------------------------------------------------------------
__global__ void cn_vnormals_kernel(const float* __restrict__ gt,
                                   const int* __restrict__ faces,
                                   float* __restrict__ vn,
                                   int B, int Ng, int F) {
    int idx = blockIdx.x * blockDim.x + threadIdx.x;
    if (idx >= B * F) return;
    int b = idx / F;
    int f = idx - b * F;
    int i0 = faces[3 * f + 0];
    int i1 = faces[3 * f + 1];
    int i2 = faces[3 * f + 2];
    const float* base = gt + (size_t)b * Ng * 3;
    float ax = base[3 * i0], ay = base[3 * i0 + 1], az = base[3 * i0 + 2];
    float bx = base[3 * i1], by = base[3 * i1 + 1], bz = base[3 * i1 + 2];
    float cx = base[3 * i2], cy = base[3 * i2 + 1], cz = base[3 * i2 + 2];
    float e1x = bx - ax, e1y = by - ay, e1z = bz - az;
    float e2x = cx - ax, e2y = cy - ay, e2z = cz - az;
    float nx = e1y * e2z - e1z * e2y;
    float ny = e1z * e2x - e1x * e2z;
    float nz = e1x * e2y - e1y * e2x;
    float* vb = vn + (size_t)b * Ng * 3;
    atomicAdd(vb + 3 * i0 + 0, nx); atomicAdd(vb + 3 * i0 + 1, ny); atomicAdd(vb + 3 * i0 + 2, nz);
    atomicAdd(vb + 3 * i1 + 0, nx); atomicAdd(vb + 3 * i1 + 1, ny); atomicAdd(vb + 3 * i1 + 2, nz);
    atomicAdd(vb + 3 * i2 + 0, nx); atomicAdd(vb + 3 * i2 + 1, ny); atomicAdd(vb + 3 * i2 + 2, nz);
}

// ---------------------------------------------------------------------------
// Kernel 2: nearest-gt-vertex per pred vertex via V_WMMA_F32_16X16X4_F32.
// One wave32 handles one (batch, 16-pred-vertex tile).
//   D = A(16 gt x 4) * B(4 x 16 pred), element = |g|^2 - 2 p.g
//   A row  (gt vertex)   = ( gx, gy, gz, |g|^2 )
//   B col  (pred vertex) = ( -2px, -2py, -2pz, 1 )
// D VGPR r, lane L -> (gt row = r + 8*(L>=16), pred col = L%16).
// Argmin: 8 INDEPENDENT (best, bestTile) chains (one per accumulator row r)
// so the per-tile compare/select ops have no serial vcc dependency; the index
// select uses the uniform tile counter (SGPR cndmask source). Chains are
// merged after the loop with full (d, gidx) lexicographic tie-break, then a
// final shfl_xor(16) merges the two lane halves -> jnp.argmin semantics.
// ---------------------------------------------------------------------------
__global__ void cn_nearest_kernel(const float* __restrict__ pred,
                                  const float* __restrict__ gt,
                                  int* __restrict__ near_idx,
                                  int B, int Np, int Ng) {
    int gtid = blockIdx.x * blockDim.x + threadIdx.x;
    int wave = gtid >> 5;
    int lane = threadIdx.x & 31;
    int ptiles = Np >> 4;                 // pred tiles of 16
    int b  = wave / ptiles;
    int pt = wave - b * ptiles;
    if (b >= B) return;                   // grid sized exactly: never taken

    int hi  = lane >> 4;                  // 0: K=0,1 half ; 1: K=2,3 half
    int l16 = lane & 15;

    // ---- B operand (fixed for the whole loop): pred vertex l16 of tile pt
    int pv = (pt << 4) + l16;
    const float* pp = pred + ((size_t)b * Np + pv) * 3;
    float px = pp[0], py = pp[1], pz = pp[2];
    v2f bmat;
    bmat.x = (hi == 0) ? (-2.0f * px) : (-2.0f * pz);   // K0 / K2
    bmat.y = (hi == 0) ? (-2.0f * py) : 1.0f;           // K1 / K3

    float best[8];
    int   bestT[8];
    #pragma unroll
    for (int r = 0; r < 8; ++r) { best[r] = 3.4e38f; bestT[r] = 0; }

    int gtiles = Ng >> 4;
    for (int t = 0; t < gtiles; ++t) {
        // ---- A operand: gt vertex l16 of tile t (both lane halves load it)
        int gvi = (t << 4) + l16;
        const float* gp = gt + ((size_t)b * Ng + gvi) * 3;
        float gx = gp[0], gy = gp[1], gz = gp[2];
        float g2 = gx * gx + gy * gy + gz * gz;
        v2f amat;
        amat.x = (hi == 0) ? gx : gz;                   // K0 / K2
        amat.y = (hi == 0) ? gy : g2;                   // K1 / K3

        v8f c = {};
        c = __builtin_amdgcn_wmma_f32_16x16x4_f32(
                /*neg_a=*/false, amat, /*neg_b=*/false, bmat,
                /*c_mod=*/(short)0, c, /*reuse_a=*/false, /*reuse_b=*/false);

        // 8 independent running-min chains; strict < keeps first (lowest t)
        #pragma unroll
        for (int r = 0; r < 8; ++r) {
            float d  = c[r];
            bool  lt = d < best[r];
            best[r]  = lt ? d : best[r];
            bestT[r] = lt ? t : bestT[r];
        }
    }

    // merge the 8 chains with (d, global index) lexicographic tie-break
    float mBest = best[0];
    int   mIdx  = (bestT[0] << 4) + (hi << 3);
    #pragma unroll
    for (int r = 1; r < 8; ++r) {
        int gidx = (bestT[r] << 4) + (hi << 3) + r;
        bool take = (best[r] < mBest) || (best[r] == mBest && gidx < mIdx);
        mBest = take ? best[r] : mBest;
        mIdx  = take ? gidx    : mIdx;
    }

    // merge lane L with lane L^16 (same pred column, other 8 gt rows)
    float obest = __shfl_xor(mBest, 16, 32);
    int   oidx  = __shfl_xor(mIdx, 16, 32);
    bool take = (obest < mBest) || (obest == mBest && oidx < mIdx);
    mBest = take ? obest : mBest;
    mIdx  = take ? oidx  : mIdx;

    if (hi == 0) near_idx[(size_t)b * Np + pv] = mIdx;
}

// ---------------------------------------------------------------------------
// Kernel 3: per-(b,edge) |dot(pred_edge, normalized near normal)|, reduce-sum
// ---------------------------------------------------------------------------
__global__ void cn_loss_kernel(const float* __restrict__ pred,
                               const int* __restrict__ edges,
                               const float* __restrict__ vn,
                               const int* __restrict__ near_idx,
                               float* __restrict__ acc,
                               int B, int Np, int Ng, int E) {
    int idx = blockIdx.x * blockDim.x + threadIdx.x;
    float val = 0.0f;
    if (idx < B * E) {
        int b = idx / E;
        int e = idx - b * E;
        int e0 = edges[2 * e + 0];
        int e1 = edges[2 * e + 1];
        const float* pb = pred + (size_t)b * Np * 3;
        float ex = pb[3 * e0 + 0] - pb[3 * e1 + 0];
        float ey = pb[3 * e0 + 1] - pb[3 * e1 + 1];
        float ez = pb[3 * e0 + 2] - pb[3 * e1 + 2];
        int ni = near_idx[(size_t)b * Np + e0];
        const float* nv = vn + ((size_t)b * Ng + ni) * 3;
        float nx = nv[0], ny = nv[1], nz = nv[2];
        float nrm = sqrtf(nx * nx + ny * ny + nz * nz);
        float inv = 1.0f / fmaxf(nrm, CN_EPS);
        val = fabsf((ex * nx + ey * ny + ez * nz) * inv);
    }
    // wave32 reduction, one atomic per wave
    #pragma unroll
    for (int off = 16; off > 0; off >>= 1)
        val += __shfl_down(val, off, 32);
    if ((threadIdx.x & 31) == 0)
        atomicAdd(acc, val);
}

// ---------------------------------------------------------------------------
// Kernel 4: finalize mean
// ---------------------------------------------------------------------------
__global__ void cn_finalize_kernel(const float* __restrict__ acc,
                                   float* __restrict__ out, float invN) {
    out[0] = acc[0] * invN;
}

// ---------------------------------------------------------------------------
extern "C" void kernel_launch(void* const* d_in, const int* in_sizes, int n_in,
                              void* d_out, int out_size, void* d_ws, size_t ws_size,
                              hipStream_t stream) {
    const float* pred  = (const float*)d_in[0];  // (B, Np, 3) f32
    const float* gt    = (const float*)d_in[1];  // (B, Ng, 3) f32
    const int*   edges = (const int*)d_in[2];    // (E, 2) int
    const int*   faces = (const int*)d_in[3];    // (F, 3) int

    const int B  = 16;
    const int Np = in_sizes[0] / (B * 3);        // 2048
    const int Ng = in_sizes[1] / (B * 3);        // 8192
    const int E  = in_sizes[2] / 2;              // directed edges
    const int F  = in_sizes[3] / 3;              // gt faces

    // workspace layout
    float* vn       = (float*)d_ws;                              // B*Ng*3 floats
    size_t vn_elems = (size_t)B * Ng * 3;
    float* acc      = vn + vn_elems;                             // 1 float
    size_t ni_off   = ((vn_elems + 4) * sizeof(float) + 15) & ~(size_t)15;
    int*   near_idx = (int*)((char*)d_ws + ni_off);              // B*Np ints

    const int TB = 256;

    // 0) zero vn + acc
    {
        int n = (int)vn_elems;
        cn_zero_kernel<<<(n + TB - 1) / TB, TB, 0, stream>>>(vn, acc, n);
    }
    // 1) vertex normals (scatter atomics)
    {
        int n = B * F;
        cn_vnormals_kernel<<<(n + TB - 1) / TB, TB, 0, stream>>>(gt, faces, vn, B, Ng, F);
    }
    // 2) nearest neighbor via WMMA: exactly B*(Np/16) waves, all fully active
    {
        int waves   = B * (Np / 16);
        int threads = waves * 32;
        cn_nearest_kernel<<<threads / TB, TB, 0, stream>>>(pred, gt, near_idx, B, Np, Ng);
    }
    // 3) edge-normal loss, summed into acc
    {
        int n = B * E;
        cn_loss_kernel<<<(n + TB - 1) / TB, TB, 0, stream>>>(pred, edges, vn, near_idx,
                                                             acc, B, Np, Ng, E);
    }
    // 4) mean
    cn_finalize_kernel<<<1, 1, 0, stream>>>(acc, (float*)d_out, 1.0f / (float)(B * E));
}